// GAT_38585986187787
// MI455X (gfx1250) — compile-verified
//
#include <hip/hip_runtime.h>
#include <hip/hip_bf16.h>

#define N_NODES 100000
#define N_EDGES 1600000
#define HC      256          // HEADS * C_OUT
#define HEADS   4
#define NEG_SLOPE 0.2f

typedef __attribute__((ext_vector_type(2))) float v2f;
typedef __attribute__((ext_vector_type(8))) float v8f;

// ---- monotone uint encoding of f32 for atomicMax-based segment max ----
__device__ __forceinline__ unsigned enc_f32(float f) {
    unsigned u = __float_as_uint(f);
    return (u & 0x80000000u) ? ~u : (u | 0x80000000u);
}
__device__ __forceinline__ float dec_f32(unsigned k) {
    unsigned u = (k & 0x80000000u) ? (k ^ 0x80000000u) : ~k;
    return __uint_as_float(u);
}
__device__ __forceinline__ float leaky(float x) {
    return x >= 0.f ? x : NEG_SLOPE * x;
}

// ---------------------------------------------------------------------
// 0) init: out = bias (broadcast per row), segmax key = 0 (== -max), denom = 0
// ---------------------------------------------------------------------
__global__ void k_init(float* __restrict__ out, const float* __restrict__ bias,
                       unsigned* __restrict__ segmax, float* __restrict__ denom) {
    int idx = blockIdx.x * blockDim.x + threadIdx.x;
    if (idx < N_NODES * HC) out[idx] = bias[idx & (HC - 1)];
    if (idx < N_NODES * HEADS) { segmax[idx] = 0u; denom[idx] = 0.f; }
}

// ---------------------------------------------------------------------
// 1) xl = x @ W via V_WMMA_F32_16X16X4_F32.
//    One wave -> 16 rows x 64 cols (4 accumulator tiles), K=64 as 16 steps.
// ---------------------------------------------------------------------
__global__ void k_gemm(const float* __restrict__ x, const float* __restrict__ W,
                       float* __restrict__ xl) {
    const int wid  = (blockIdx.x * blockDim.x + threadIdx.x) >> 5; // 0..24999
    const int lane = threadIdx.x & 31;
    const int mt = wid >> 2;          // 0..6249 : row tile
    const int qt = wid & 3;           // 0..3    : 64-col quarter (== head)
    const int m0 = mt * 16;
    const int hi = lane >> 4;         // half-wave id (0/1)
    const int l16 = lane & 15;

    v8f acc0 = {}, acc1 = {}, acc2 = {}, acc3 = {};

    const float* xrow  = x + (size_t)(m0 + l16) * 64 + 2 * hi;      // A fragment base
    const float* wbase = W + (size_t)(2 * hi) * HC + qt * 64 + l16; // B fragment base

#pragma unroll
    for (int kk = 0; kk < 16; ++kk) {
        const int kb = kk * 4;
        // A 16x4 f32: reg r holds K = kb + r + 2*hi  -> contiguous float2
        v2f a = *(const v2f*)(xrow + kb);
        const float* wp = wbase + (size_t)kb * HC;
        // B 4x16 f32: reg r holds K = kb + r + 2*hi, N = l16
        v2f b0 = { wp[0],  wp[0  + HC] };
        v2f b1 = { wp[16], wp[16 + HC] };
        v2f b2 = { wp[32], wp[32 + HC] };
        v2f b3 = { wp[48], wp[48 + HC] };
        acc0 = __builtin_amdgcn_wmma_f32_16x16x4_f32(false, a, false, b0, (short)0, acc0, false, false);
        acc1 = __builtin_amdgcn_wmma_f32_16x16x4_f32(false, a, false, b1, (short)0, acc1, false, false);
        acc2 = __builtin_amdgcn_wmma_f32_16x16x4_f32(false, a, false, b2, (short)0, acc2, false, false);
        acc3 = __builtin_amdgcn_wmma_f32_16x16x4_f32(false, a, false, b3, (short)0, acc3, false, false);
    }

    // C/D layout: reg r -> (M = m0 + r + 8*hi, N = l16)
    float* orow = xl + (size_t)(m0 + 8 * hi) * HC + qt * 64 + l16;
#pragma unroll
    for (int r = 0; r < 8; ++r) {
        orow[(size_t)r * HC + 0]  = acc0[r];
        orow[(size_t)r * HC + 16] = acc1[r];
        orow[(size_t)r * HC + 32] = acc2[r];
        orow[(size_t)r * HC + 48] = acc3[r];
    }
}

// ---------------------------------------------------------------------
// 2) per-(node,head) attention scalars: wave-per-(n,h) dot products
// ---------------------------------------------------------------------
__global__ void k_attn(const float* __restrict__ xl,
                       const float* __restrict__ att_src, const float* __restrict__ att_dst,
                       float* __restrict__ a_src, float* __restrict__ a_dst) {
    const int wid  = (blockIdx.x * blockDim.x + threadIdx.x) >> 5;
    const int lane = threadIdx.x & 31;
    if (wid >= N_NODES * HEADS) return;
    const int n = wid >> 2, h = wid & 3;
    const float* xp = xl + (size_t)n * HC + h * 64;
    const float* as = att_src + h * 64;
    const float* ad = att_dst + h * 64;
    float v0 = xp[lane], v1 = xp[lane + 32];
    float s = v0 * as[lane] + v1 * as[lane + 32];
    float d = v0 * ad[lane] + v1 * ad[lane + 32];
#pragma unroll
    for (int m = 16; m >= 1; m >>= 1) {
        s += __shfl_xor(s, m, 32);
        d += __shfl_xor(d, m, 32);
    }
    if (lane == 0) { a_src[wid] = s; a_dst[wid] = d; }
}

// ---------------------------------------------------------------------
// 3) segment max of leaky-relu logits over destination nodes
// ---------------------------------------------------------------------
__global__ void k_segmax(const long long* __restrict__ ei,
                         const float* __restrict__ a_src, const float* __restrict__ a_dst,
                         unsigned* __restrict__ segmax) {
    int t = blockIdx.x * blockDim.x + threadIdx.x;
    if (t >= N_EDGES * HEADS) return;
    const int e = t >> 2, h = t & 3;
    const int s = (int)ei[e];
    const int d = (int)ei[N_EDGES + e];
    float lg = leaky(a_src[s * HEADS + h] + a_dst[d * HEADS + h]);
    atomicMax(segmax + d * HEADS + h, enc_f32(lg));
}

// ---------------------------------------------------------------------
// 4) denom = segment_sum(exp(logit - segmax[dst]))
// ---------------------------------------------------------------------
__global__ void k_denom(const long long* __restrict__ ei,
                        const float* __restrict__ a_src, const float* __restrict__ a_dst,
                        const unsigned* __restrict__ segmax, float* __restrict__ denom) {
    int t = blockIdx.x * blockDim.x + threadIdx.x;
    if (t >= N_EDGES * HEADS) return;
    const int e = t >> 2, h = t & 3;
    const int s = (int)ei[e];
    const int d = (int)ei[N_EDGES + e];
    float lg = leaky(a_src[s * HEADS + h] + a_dst[d * HEADS + h]);
    float ev = __expf(lg - dec_f32(segmax[d * HEADS + h]));
    atomicAdd(denom + d * HEADS + h, ev);
}

// ---------------------------------------------------------------------
// 5) weighted scatter: wave per edge; lanes 0..3 compute head alphas,
//    broadcast, then 8 coalesced 128B sweeps of gather*alpha + atomicAdd
// ---------------------------------------------------------------------
__global__ void k_scatter(const long long* __restrict__ ei,
                          const float* __restrict__ xl,
                          const float* __restrict__ a_src, const float* __restrict__ a_dst,
                          const unsigned* __restrict__ segmax, const float* __restrict__ denom,
                          float* __restrict__ out) {
    const int wid  = (blockIdx.x * blockDim.x + threadIdx.x) >> 5;
    const int lane = threadIdx.x & 31;
    if (wid >= N_EDGES) return;
    const int s = (int)ei[wid];
    const int d = (int)ei[N_EDGES + wid];

    float alpha = 0.f;
    if (lane < HEADS) {
        float lg = leaky(a_src[s * HEADS + lane] + a_dst[d * HEADS + lane]);
        float ev = __expf(lg - dec_f32(segmax[d * HEADS + lane]));
        alpha = ev / (denom[d * HEADS + lane] + 1e-16f);
    }
    const float al0 = __shfl(alpha, 0, 32);
    const float al1 = __shfl(alpha, 1, 32);
    const float al2 = __shfl(alpha, 2, 32);
    const float al3 = __shfl(alpha, 3, 32);

    const float* xr = xl + (size_t)s * HC;
    float* orow = out + (size_t)d * HC;
#pragma unroll
    for (int j = 0; j < 8; ++j) {                  // c = lane + 32*j, head = j>>1
        const float a = (j < 2) ? al0 : (j < 4) ? al1 : (j < 6) ? al2 : al3;
        const int c = lane + 32 * j;
        atomicAdd(&orow[c], xr[c] * a);
    }
}

extern "C" void kernel_launch(void* const* d_in, const int* in_sizes, int n_in,
                              void* d_out, int out_size, void* d_ws, size_t ws_size,
                              hipStream_t stream) {
    const float*     x       = (const float*)d_in[0];
    const long long* ei      = (const long long*)d_in[1];   // int64 edge_index [2, E]
    const float*     W       = (const float*)d_in[2];
    const float*     att_src = (const float*)d_in[3];
    const float*     att_dst = (const float*)d_in[4];
    const float*     bias    = (const float*)d_in[5];
    float*           out     = (float*)d_out;

    // workspace carve-up (~109 MB)
    float*    xl     = (float*)d_ws;                            // N*256
    float*    a_src  = xl + (size_t)N_NODES * HC;               // N*4
    float*    a_dst  = a_src + (size_t)N_NODES * HEADS;         // N*4
    unsigned* segmax = (unsigned*)(a_dst + (size_t)N_NODES * HEADS);
    float*    denom  = (float*)(segmax + (size_t)N_NODES * HEADS);

    const int B = 256;
    // 0) init out=bias, segmax/denom=0
    k_init<<<(N_NODES * HC + B - 1) / B, B, 0, stream>>>(out, bias, segmax, denom);
    // 1) GEMM: 6250*4 waves = 25000 waves -> 3125 blocks of 8 waves
    k_gemm<<<3125, B, 0, stream>>>(x, W, xl);
    // 2) attention scalars: N*H waves
    k_attn<<<(N_NODES * HEADS * 32 + B - 1) / B, B, 0, stream>>>(xl, att_src, att_dst, a_src, a_dst);
    // 3) segment max
    k_segmax<<<(N_EDGES * HEADS + B - 1) / B, B, 0, stream>>>(ei, a_src, a_dst, segmax);
    // 4) denom
    k_denom<<<(N_EDGES * HEADS + B - 1) / B, B, 0, stream>>>(ei, a_src, a_dst, segmax, denom);
    // 5) scatter: one wave per edge
    k_scatter<<<(N_EDGES * 32 + B - 1) / B, B, 0, stream>>>(ei, xl, a_src, a_dst, segmax, denom, out);
}